// GCNModel_15401752723911
// MI455X (gfx1250) — compile-verified
//
#include <hip/hip_runtime.h>
#include <stdint.h>

typedef float v2f __attribute__((ext_vector_type(2)));
typedef float v8f __attribute__((ext_vector_type(8)));

#define N_NODES   50000
#define N_EDGES   800000
#define IN_DIM    128
#define HID_DIM   128
#define N_CLASSES 64

// ---------------- degree / norm kernels ----------------

__global__ __launch_bounds__(256) void k_deg_init(float* __restrict__ deg) {
  int i = blockIdx.x * 256 + threadIdx.x;
  if (i < N_NODES) deg[i] = 1.0f;  // self-loop
}

__global__ __launch_bounds__(256) void k_deg_count(const int* __restrict__ dst,
                                                   float* __restrict__ deg) {
  int e = blockIdx.x * 256 + threadIdx.x;
  if (e < N_EDGES) atomicAdd(&deg[dst[e]], 1.0f);
}

__global__ __launch_bounds__(256) void k_dinv(float* __restrict__ deg) {
  int i = blockIdx.x * 256 + threadIdx.x;
  if (i < N_NODES) deg[i] = rsqrtf(deg[i]);  // deg >= 1 always (self-loop)
}

// ---------------- WMMA GEMM: C[M,NT*16] = (relu?)A[M,128] * B[128,NT*16] ----
// One wave computes a 16-row M tile across all NT 16-col tiles.
// 8 waves per block -> 128 rows per block.
// B is staged in LDS *pre-swizzled by k-pairs*: Bs[kp*NOUT+n] = {B[2kp][n],
// B[2kp+1][n]} so each lane's fragment is one aligned ds_load_b64 landing in
// an even VGPR pair (WMMA SRC1 requirement) -- no repacking movs.

template <int NT, bool RELU>
__global__ __launch_bounds__(256) void k_gemm(const float* __restrict__ A,
                                              const float* __restrict__ B,
                                              float* __restrict__ C, int M) {
  constexpr int K = 128;
  constexpr int NOUT = NT * 16;
  constexpr int NSHIFT = (NT == 8) ? 7 : 6;   // log2(NOUT)
  __shared__ v2f Bs[(K / 2) * NOUT];

  // cooperative swizzled stage of B into LDS
  for (int i = threadIdx.x; i < (K / 2) * NOUT; i += 256) {
    const int kp = i >> NSHIFT;
    const int n  = i & (NOUT - 1);
    v2f v;
    v.x = B[(2 * kp    ) * NOUT + n];
    v.y = B[(2 * kp + 1) * NOUT + n];
    Bs[i] = v;
  }
  __syncthreads();

  const int wave = threadIdx.x >> 5;
  const int lane = threadIdx.x & 31;
  const int half = lane >> 4;   // 0: lanes 0-15, 1: lanes 16-31
  const int l16  = lane & 15;
  const int mbase = (blockIdx.x * 8 + wave) * 16;

  v8f acc[NT];
  #pragma unroll
  for (int t = 0; t < NT; ++t) acc[t] = (v8f){0.f,0.f,0.f,0.f,0.f,0.f,0.f,0.f};

  // clamp (not predicate) the tail rows: EXEC must stay all-1s for WMMA
  int arow = mbase + l16;
  if (arow >= M) arow = M - 1;
  const float* Arow = A + (size_t)arow * K;

  #pragma unroll 4
  for (int kk = 0; kk < K; kk += 4) {
    // lanes 0-15 consume K=kk,kk+1 (pair kp) ; lanes 16-31 K=kk+2,kk+3 (kp+1)
    const int kp = (kk >> 1) + half;
    v2f a = *(const v2f*)(Arow + kk + half * 2);
    if (RELU) { a.x = fmaxf(a.x, 0.f); a.y = fmaxf(a.y, 0.f); }
    const v2f* brow = Bs + kp * NOUT + l16;
    #pragma unroll
    for (int t = 0; t < NT; ++t) {
      v2f b = brow[t * 16];   // single aligned ds_load_b64, imm offset t*128B
      acc[t] = __builtin_amdgcn_wmma_f32_16x16x4_f32(
          /*neg_a=*/false, a, /*neg_b=*/false, b,
          /*c_mod=*/(short)0, acc[t], /*reuse_a=*/false, /*reuse_b=*/false);
    }
  }

  // C/D layout: VGPR r holds M=r (lanes 0-15) and M=r+8 (lanes 16-31), N=l16
  #pragma unroll
  for (int t = 0; t < NT; ++t) {
    #pragma unroll
    for (int r = 0; r < 8; ++r) {
      const int row = mbase + r + half * 8;
      if (row < M) C[(size_t)row * NOUT + t * 16 + l16] = acc[t][r];
    }
  }
}

// ---------------- self-loop + bias init: OUT = H*dinv^2 + bias --------------

template <int F>
__global__ __launch_bounds__(256) void k_selfinit(const float* __restrict__ H,
                                                  const float* __restrict__ dinv,
                                                  const float* __restrict__ bias,
                                                  float* __restrict__ OUT) {
  int i = blockIdx.x * 256 + threadIdx.x;
  if (i >= N_NODES * F) return;
  int node = i / F;
  int f    = i & (F - 1);
  float dv = dinv[node];
  OUT[i] = H[i] * dv * dv + bias[f];
}

// ---------------- edge scatter: OUT[dst] += H[src] * dinv[src]*dinv[dst] ----
// one thread per (edge, float4 chunk); consecutive lanes cover one feature row

template <int F>
__global__ __launch_bounds__(256) void k_scatter(const int* __restrict__ src,
                                                 const int* __restrict__ dst,
                                                 const float* __restrict__ dinv,
                                                 const float* __restrict__ H,
                                                 float* __restrict__ OUT) {
  constexpr int CH = F / 4;
  int idx = blockIdx.x * 256 + threadIdx.x;
  if (idx >= N_EDGES * CH) return;
  int e = idx / CH;
  int c = idx & (CH - 1);
  int s = src[e];
  int d = dst[e];
  float nrm = dinv[s] * dinv[d];
  float4 h = ((const float4*)(H + (size_t)s * F))[c];
  float* op = OUT + (size_t)d * F + c * 4;
  atomicAdd(op + 0, h.x * nrm);
  atomicAdd(op + 1, h.y * nrm);
  atomicAdd(op + 2, h.z * nrm);
  atomicAdd(op + 3, h.w * nrm);
}

// ---------------- launch ----------------------------------------------------

extern "C" void kernel_launch(void* const* d_in, const int* in_sizes, int n_in,
                              void* d_out, int out_size, void* d_ws, size_t ws_size,
                              hipStream_t stream) {
  (void)in_sizes; (void)n_in; (void)out_size; (void)ws_size;

  const float* x    = (const float*)d_in[0];          // [50000,128]
  const int*   ei   = (const int*)d_in[1];            // [2,800000] (int per harness)
  const float* W1   = (const float*)d_in[2];          // [128,128]
  const float* b1   = (const float*)d_in[3];          // [128]
  const float* W2   = (const float*)d_in[4];          // [128,64]
  const float* b2   = (const float*)d_in[5];          // [64]
  const int* src = ei;
  const int* dst = ei + N_EDGES;

  // workspace carve-up (H2 reuses H1's slot; never live together)
  char* ws = (char*)d_ws;
  float* dinv = (float*)(ws + 0);                         //  50000 f
  float* H1   = (float*)(ws + 200192);                    //  50000*128 f (25.6MB)
  float* AGG  = (float*)(ws + 200192 + 25600000);         //  50000*128 f (25.6MB)
  float* H2   = H1;                                       //  50000*64 f, reuse

  float* OUT = (float*)d_out;                             //  50000*64 f

  const int nb_nodes  = (N_NODES + 255) / 256;
  const int nb_edges  = (N_EDGES + 255) / 256;
  const int nb_gemm   = (N_NODES + 127) / 128;            // 391
  const int nb_nf128  = (N_NODES * 128 + 255) / 256;
  const int nb_nf64   = (N_NODES * 64  + 255) / 256;
  const int nb_sc128  = (N_EDGES * 32 + 255) / 256;       // 100000
  const int nb_sc64   = (N_EDGES * 16 + 255) / 256;       // 50000

  // degrees -> dinv (in place)
  k_deg_init <<<nb_nodes, 256, 0, stream>>>(dinv);
  k_deg_count<<<nb_edges, 256, 0, stream>>>(dst, dinv);
  k_dinv     <<<nb_nodes, 256, 0, stream>>>(dinv);

  // layer 1
  k_gemm<8, false><<<nb_gemm, 256, 0, stream>>>(x, W1, H1, N_NODES);
  k_selfinit<128> <<<nb_nf128, 256, 0, stream>>>(H1, dinv, b1, AGG);
  k_scatter<128>  <<<nb_sc128, 256, 0, stream>>>(src, dst, dinv, H1, AGG);

  // layer 2 (ReLU fused into GEMM2 A-fragment load)
  k_gemm<4, true> <<<nb_gemm, 256, 0, stream>>>(AGG, W2, H2, N_NODES);
  k_selfinit<64>  <<<nb_nf64, 256, 0, stream>>>(H2, dinv, b2, OUT);
  k_scatter<64>   <<<nb_sc64, 256, 0, stream>>>(src, dst, dinv, H2, OUT);
}